// CustomMultiheadAttention_63333587747277
// MI455X (gfx1250) — compile-verified
//
#include <hip/hip_runtime.h>

// Problem constants (from reference): S=2048, B=2, E=768, H=12, D=64
#define S_LEN 2048
#define BATCH 2
#define EMB   768
#define NHEAD 12
#define HDIM  64

typedef __attribute__((ext_vector_type(2))) float v2f;
typedef __attribute__((ext_vector_type(8))) float v8f;
typedef int v4i_gcc __attribute__((vector_size(4 * sizeof(int))));  // matches builtin pointee

#define BM 64
#define BN 64
#define BK 32
#define PAD_A 36   // 32 + 4: float4-aligned, conflict-free fragment reads
#define PAD_B 68   // 64 + 4: float4-aligned (NN B tile)

// ---------------------------------------------------------------------------
// CDNA5 async global->LDS staging (ASYNCcnt-tracked), with sync fallback.
// ---------------------------------------------------------------------------
#if __has_builtin(__builtin_amdgcn_global_load_async_to_lds_b128) && \
    __has_builtin(__builtin_amdgcn_s_wait_asynccnt)
#define HAVE_ASYNC_LDS 1
#else
#define HAVE_ASYNC_LDS 0
#endif

__device__ __forceinline__ void copy16_to_lds(const float* g, float* l)
{
#if HAVE_ASYNC_LDS
    // LDS flat-address low 32 bits are the LDS byte offset (aperture rule),
    // so integer-truncating the generic pointer yields a valid AS(3) pointer.
    __builtin_amdgcn_global_load_async_to_lds_b128(
        (v4i_gcc __attribute__((address_space(1)))*)(uintptr_t)g,
        (v4i_gcc __attribute__((address_space(3)))*)(uint32_t)(uintptr_t)l,
        0, 0);
#else
    *(float4*)l = *(const float4*)g;
#endif
}

#if HAVE_ASYNC_LDS
#define ASYNC_WAIT(n) __builtin_amdgcn_s_wait_asynccnt(n)
#else
#define ASYNC_WAIT(n) ((void)0)
#endif

// ---------------------------------------------------------------------------
// NT GEMM: C[m,n] = scale * sum_k A[m,k]*B[n,k] (+ bias[n])
//   A_MODE 0: A row-major with lda
//   A_MODE 1: A gathered from ctx layout [B*H, S, D]: (m,k) ->
//             ctx[((m&1)*H + k/64)*S*D + (m>>1)*D + (k&63)]
//   C_MODE 0: row-major C with ldc (non-temporal stores: write-once streams)
//   C_MODE 1: head-split scatter (regular stores: consumed by next kernel)
// Batched via blockIdx.z with 64-bit strides. Double-buffered LDS pipeline.
// ---------------------------------------------------------------------------
template<int A_MODE, int C_MODE, bool HAS_BIAS>
__global__ void __launch_bounds__(256)
wmma_gemm_nt(const float* __restrict__ A, int lda, long long strideA,
             const float* __restrict__ B, int ldb, long long strideB,
             const float* __restrict__ bias,
             float* __restrict__ C, int ldc, long long strideC,
             int K, float scale)
{
    __shared__ float As[2][BM][PAD_A];
    __shared__ float Bs[2][BN][PAD_A];

    const int z = blockIdx.z;
    const float* Ab = A + (size_t)z * (size_t)strideA;
    const float* Bb = B + (size_t)z * (size_t)strideB;
    float*       Cb = C + (size_t)z * (size_t)strideC;

    const int m0 = blockIdx.x * BM;
    const int n0 = blockIdx.y * BN;

    const int tid  = threadIdx.x;
    const int wave = tid >> 5;
    const int lane = tid & 31;
    const int hi   = lane >> 4;     // 0: lanes 0-15, 1: lanes 16-31
    const int lr   = lane & 15;
    const int m_sub = (wave & 1) * 32;   // each wave owns 32x16 (two 16x16 tiles)
    const int n_sub = (wave >> 1) * 16;

    v8f acc0 = {};
    v8f acc1 = {};

    // 4 x 16B transfers per thread per tile (2 for A, 2 for B)
    auto stageTile = [&](int k0, int buf) {
#pragma unroll
        for (int i = 0; i < 2; ++i) {
            const int linear = tid + i * 256;      // 0..511 float4 slots
            const int row = linear >> 3;           // 8 float4 per 32-float row
            const int c4  = (linear & 7) * 4;
            const float* ap;
            if (A_MODE == 0) {
                ap = Ab + (size_t)(m0 + row) * (size_t)lda + (size_t)(k0 + c4);
            } else {
                const int m = m0 + row;
                const int head = k0 >> 6;          // BK=32 never straddles a head
                ap = Ab + (((size_t)(m & 1) * NHEAD + head) * S_LEN + (size_t)(m >> 1)) * HDIM
                        + (size_t)((k0 & 63) + c4);
            }
            copy16_to_lds(ap, &As[buf][row][c4]);
            copy16_to_lds(Bb + (size_t)(n0 + row) * (size_t)ldb + (size_t)(k0 + c4),
                          &Bs[buf][row][c4]);
        }
    };

    const int nTiles = K / BK;
    stageTile(0, 0);

    for (int t = 0; t < nTiles; ++t) {
        const int buf = t & 1;
        if (t + 1 < nTiles) {
            stageTile((t + 1) * BK, buf ^ 1);
            ASYNC_WAIT(4);          // in-order completion => tile t has landed
        } else {
            ASYNC_WAIT(0);
        }
        __syncthreads();            // everyone's tile-t data visible

#pragma unroll
        for (int kk = 0; kk < BK; kk += 4) {
            const int ka = kk + (hi << 1);   // lanes 16-31 hold K=2,3
            v2f a0, a1, bf;
            a0.x = As[buf][m_sub + lr][ka];        a0.y = As[buf][m_sub + lr][ka + 1];
            a1.x = As[buf][m_sub + 16 + lr][ka];   a1.y = As[buf][m_sub + 16 + lr][ka + 1];
            bf.x = Bs[buf][n_sub + lr][ka];        bf.y = Bs[buf][n_sub + lr][ka + 1];
            acc0 = __builtin_amdgcn_wmma_f32_16x16x4_f32(false, a0, false, bf,
                                                         (short)0, acc0, false, false);
            acc1 = __builtin_amdgcn_wmma_f32_16x16x4_f32(false, a1, false, bf,
                                                         (short)0, acc1, false, false);
        }
        __syncthreads();            // all waves done reading buf before refill
    }

    // --- epilogue: C/D layout: VGPR i, lanes0-15 -> M=i, lanes16-31 -> M=i+8, N=lr
    const int nG = n0 + n_sub + lr;
    const float bval = HAS_BIAS ? bias[nG] : 0.0f;
#pragma unroll
    for (int i = 0; i < 8; ++i) {
        const int mG0 = m0 + m_sub + hi * 8 + i;
        const int mG1 = mG0 + 16;
        const float v0 = acc0[i] * scale + bval;
        const float v1 = acc1[i] * scale + bval;
        if (C_MODE == 0) {
            __builtin_nontemporal_store(v0, &Cb[(size_t)mG0 * (size_t)ldc + nG]);
            __builtin_nontemporal_store(v1, &Cb[(size_t)mG1 * (size_t)ldc + nG]);
        } else {
            const int h = nG >> 6, d = nG & 63;
            Cb[(((size_t)(mG0 & 1) * NHEAD + h) * S_LEN + (size_t)(mG0 >> 1)) * HDIM + d] = v0;
            Cb[(((size_t)(mG1 & 1) * NHEAD + h) * S_LEN + (size_t)(mG1 >> 1)) * HDIM + d] = v1;
        }
    }
}

// ---------------------------------------------------------------------------
// NN GEMM: C[m,n] = sum_k A[m,k]*B[k,n]   (used for ctx = P @ V)
// ---------------------------------------------------------------------------
__global__ void __launch_bounds__(256)
wmma_gemm_nn(const float* __restrict__ A, int lda, long long strideA,
             const float* __restrict__ B, int ldb, long long strideB,
             float* __restrict__ C, int ldc, long long strideC,
             int K)
{
    __shared__ float As[2][BM][PAD_A];
    __shared__ float Bs[2][BK][PAD_B];

    const int z = blockIdx.z;
    const float* Ab = A + (size_t)z * (size_t)strideA;
    const float* Bb = B + (size_t)z * (size_t)strideB;
    float*       Cb = C + (size_t)z * (size_t)strideC;

    const int m0 = blockIdx.x * BM;
    const int n0 = blockIdx.y * BN;

    const int tid  = threadIdx.x;
    const int wave = tid >> 5;
    const int lane = tid & 31;
    const int hi   = lane >> 4;
    const int lr   = lane & 15;
    const int m_sub = (wave & 1) * 32;
    const int n_sub = (wave >> 1) * 16;

    v8f acc0 = {};
    v8f acc1 = {};

    auto stageTile = [&](int k0, int buf) {
#pragma unroll
        for (int i = 0; i < 2; ++i) {
            const int linear = tid + i * 256;
            { // A tile: BM x BK (attn probabilities, streamed)
                const int row = linear >> 3;
                const int c4  = (linear & 7) * 4;
                copy16_to_lds(Ab + (size_t)(m0 + row) * (size_t)lda + (size_t)(k0 + c4),
                              &As[buf][row][c4]);
            }
            { // B tile: BK x BN
                const int row = linear >> 4;       // 16 float4 per 64-float row
                const int c4  = (linear & 15) * 4;
                copy16_to_lds(Bb + (size_t)(k0 + row) * (size_t)ldb + (size_t)(n0 + c4),
                              &Bs[buf][row][c4]);
            }
        }
    };

    const int nTiles = K / BK;
    stageTile(0, 0);

    for (int t = 0; t < nTiles; ++t) {
        const int buf = t & 1;
        if (t + 1 < nTiles) {
            stageTile((t + 1) * BK, buf ^ 1);
            ASYNC_WAIT(4);
        } else {
            ASYNC_WAIT(0);
        }
        __syncthreads();

#pragma unroll
        for (int kk = 0; kk < BK; kk += 4) {
            const int ka = kk + (hi << 1);
            v2f a0, a1, bf;
            a0.x = As[buf][m_sub + lr][ka];        a0.y = As[buf][m_sub + lr][ka + 1];
            a1.x = As[buf][m_sub + 16 + lr][ka];   a1.y = As[buf][m_sub + 16 + lr][ka + 1];
            bf.x = Bs[buf][ka][n_sub + lr];        bf.y = Bs[buf][ka + 1][n_sub + lr];
            acc0 = __builtin_amdgcn_wmma_f32_16x16x4_f32(false, a0, false, bf,
                                                         (short)0, acc0, false, false);
            acc1 = __builtin_amdgcn_wmma_f32_16x16x4_f32(false, a1, false, bf,
                                                         (short)0, acc1, false, false);
        }
        __syncthreads();
    }

    const int nG = n0 + n_sub + lr;
#pragma unroll
    for (int i = 0; i < 8; ++i) {
        const int mG0 = m0 + m_sub + hi * 8 + i;
        Cb[(size_t)mG0 * (size_t)ldc + nG]        = acc0[i];   // ctx: re-read by K5
        Cb[(size_t)(mG0 + 16) * (size_t)ldc + nG] = acc1[i];
    }
}

// ---------------------------------------------------------------------------
// Row softmax over 2048 columns, in place. One 256-thread block per row.
// wave32 shuffle reductions + 8-entry LDS hop for the cross-wave step.
// ---------------------------------------------------------------------------
__global__ void __launch_bounds__(256)
softmax_rows(float* __restrict__ attn)
{
    __shared__ float red[8];
    float* p = attn + (size_t)blockIdx.x * S_LEN;
    const int tid  = threadIdx.x;
    const int wave = tid >> 5;
    const int lane = tid & 31;

    float vals[8];
    float mx = -3.402823466e38f;
#pragma unroll
    for (int i = 0; i < 8; ++i) {
        vals[i] = p[tid + i * 256];
        mx = fmaxf(mx, vals[i]);
    }
#pragma unroll
    for (int off = 16; off > 0; off >>= 1) mx = fmaxf(mx, __shfl_xor(mx, off, 32));
    if (lane == 0) red[wave] = mx;
    __syncthreads();
    mx = red[0];
#pragma unroll
    for (int w = 1; w < 8; ++w) mx = fmaxf(mx, red[w]);

    float sum = 0.0f;
#pragma unroll
    for (int i = 0; i < 8; ++i) {
        vals[i] = __expf(vals[i] - mx);
        sum += vals[i];
    }
#pragma unroll
    for (int off = 16; off > 0; off >>= 1) sum += __shfl_xor(sum, off, 32);
    __syncthreads();                 // done reading red[] for the max step
    if (lane == 0) red[wave] = sum;
    __syncthreads();
    sum = red[0];
#pragma unroll
    for (int w = 1; w < 8; ++w) sum += red[w];

    const float inv = 1.0f / sum;
#pragma unroll
    for (int i = 0; i < 8; ++i) p[tid + i * 256] = vals[i] * inv;
}

// ---------------------------------------------------------------------------
extern "C" void kernel_launch(void* const* d_in, const int* in_sizes, int n_in,
                              void* d_out, int out_size, void* d_ws, size_t ws_size,
                              hipStream_t stream)
{
    const float* query = (const float*)d_in[0];
    const float* key_  = (const float*)d_in[1];
    const float* value = (const float*)d_in[2];
    const float* W_Q   = (const float*)d_in[3];
    const float* W_K   = (const float*)d_in[4];
    const float* W_V   = (const float*)d_in[5];
    const float* b_Q   = (const float*)d_in[6];
    const float* b_K   = (const float*)d_in[7];
    const float* b_V   = (const float*)d_in[8];
    const float* W_O   = (const float*)d_in[9];
    const float* b_O   = (const float*)d_in[10];

    float* out  = (float*)d_out;                               // [S, B, E]
    float* attn = out + (size_t)S_LEN * BATCH * EMB;           // [B*H, S, S]

    const size_t headBuf = (size_t)BATCH * NHEAD * S_LEN * HDIM;  // 3,145,728 floats
    float* Qh  = (float*)d_ws;
    float* Kh  = Qh + headBuf;
    float* Vh  = Kh + headBuf;
    float* ctx = Vh + headBuf;                                 // total 48 MB

    const long long hs  = (long long)S_LEN * HDIM;             // per-head Q/K/V/ctx stride
    const long long as_ = (long long)S_LEN * S_LEN;            // per-head attn stride

    const dim3 blk(256);

    // 1) QKV projections: [4096,768] x [768,768]^T + bias -> head-split layout
    const dim3 gP(S_LEN * BATCH / BM, EMB / BN, 1);
    hipLaunchKernelGGL((wmma_gemm_nt<0, 1, true>), gP, blk, 0, stream,
                       query, EMB, 0LL, W_Q, EMB, 0LL, b_Q, Qh, 0, 0LL, EMB, 1.0f);
    hipLaunchKernelGGL((wmma_gemm_nt<0, 1, true>), gP, blk, 0, stream,
                       key_,  EMB, 0LL, W_K, EMB, 0LL, b_K, Kh, 0, 0LL, EMB, 1.0f);
    hipLaunchKernelGGL((wmma_gemm_nt<0, 1, true>), gP, blk, 0, stream,
                       value, EMB, 0LL, W_V, EMB, 0LL, b_V, Vh, 0, 0LL, EMB, 1.0f);

    // 2) scores = Q K^T / sqrt(D), batched over 24 heads (NT stores: 402 MB stream)
    const dim3 gS(S_LEN / BM, S_LEN / BN, BATCH * NHEAD);
    hipLaunchKernelGGL((wmma_gemm_nt<0, 0, false>), gS, blk, 0, stream,
                       Qh, HDIM, hs, Kh, HDIM, hs, (const float*)nullptr,
                       attn, S_LEN, as_, HDIM, 0.125f);

    // 3) row softmax in place on attn (this IS the second reference output)
    hipLaunchKernelGGL(softmax_rows, dim3(BATCH * NHEAD * S_LEN), blk, 0, stream, attn);

    // 4) ctx = P @ V, batched over 24 heads
    const dim3 gC(S_LEN / BM, HDIM / BN, BATCH * NHEAD);
    hipLaunchKernelGGL(wmma_gemm_nn, gC, blk, 0, stream,
                       attn, S_LEN, as_, Vh, HDIM, hs, ctx, HDIM, hs, S_LEN);

    // 5) output = merge_heads(ctx) @ W_O^T + b_O  (A gathered from ctx layout)
    const dim3 gO(S_LEN * BATCH / BM, EMB / BN, 1);
    hipLaunchKernelGGL((wmma_gemm_nt<1, 0, true>), gO, blk, 0, stream,
                       ctx, 0, 0LL, W_O, EMB, 0LL, b_O, out, EMB, 0LL, EMB, 1.0f);

    (void)in_sizes; (void)n_in; (void)out_size; (void)ws_size;
}